// GaussianScene_66683662238114
// MI455X (gfx1250) — compile-verified
//
#include <hip/hip_runtime.h>
#include <stdint.h>

typedef unsigned int u32;

#define RADIX 16
#define SORT_TILE 4096
#define SORT_THREADS 256
#define SORT_ITEMS 16

typedef unsigned int v4u __attribute__((ext_vector_type(4)));
typedef int v8i __attribute__((ext_vector_type(8)));
typedef int v4i __attribute__((ext_vector_type(4)));

#if defined(__has_builtin)
#  if __has_builtin(__builtin_amdgcn_tensor_load_to_lds) && __has_builtin(__builtin_amdgcn_s_wait_tensorcnt)
#    define USE_TDM 1
#  else
#    define USE_TDM 0
#  endif
#else
#  define USE_TDM 0
#endif

// Monotone float -> u32 transform (all keys here are positive or +inf, but be general).
__device__ __forceinline__ u32 key_bits(float f) {
  u32 b = __float_as_uint(f);
  return b ^ ((u32)((int)b >> 31) | 0x80000000u);
}

#if USE_TDM
// Tensor Data Mover: DMA a contiguous run of DWORDs into LDS.
// D# per CDNA5 ISA: count=1, type=2(image), data_size=4B,
// 1D tile of tile_dwords elements; tensor_dim0 = avail_dwords so the TDM
// zero-fills out-of-bounds reads on the last (partial) tile.
__device__ __forceinline__ void tdm_load_dwords(void* lds_dst, const void* gsrc,
                                                u32 avail_dwords, u32 tile_dwords) {
  const uint64_t ga = (uint64_t)(uintptr_t)gsrc;
  v4u g0;
  g0.x = 1u;                                              // count=1 (valid descriptor)
  g0.y = (u32)(uintptr_t)lds_dst;                         // LDS destination address (low 32 bits)
  g0.z = (u32)ga;                                         // global_addr[31:0]
  g0.w = ((u32)(ga >> 32) & 0x01FFFFFFu) | (2u << 30);    // global_addr[56:32] | type=2
  v8i g1;
  g1[0] = (int)(2u << 16);                                // data_size=2 -> 4 bytes
  g1[1] = (int)((avail_dwords & 0xFFFFu) << 16);          // tensor_dim0[15:0]  @ [63:48]
  g1[2] = (int)((avail_dwords >> 16) | (1u << 16));       // tensor_dim0[31:16] | tensor_dim1=1
  g1[3] = (int)(tile_dwords << 16);                       // tile_dim0 @ [127:112]
  g1[4] = 0;                                              // tile_dim1=0, tile_dim2=0
  g1[5] = (int)avail_dwords;                              // tensor_dim0_stride[31:0]
  g1[6] = 0;                                              // stride[47:32]=0
  g1[7] = 0;
  v4i gz = {0, 0, 0, 0};                                  // groups 2/3 unused (<=2D)
  v8i gz8 = {0, 0, 0, 0, 0, 0, 0, 0};
  __builtin_amdgcn_tensor_load_to_lds(g0, g1, gz, gz, gz8, 0);
}
#endif

// -------------------------------------------------------------------------
// Pass 1: depth keys, point tile staged into LDS by the TDM.
// -------------------------------------------------------------------------
__global__ void gs_keys_kernel(const float* __restrict__ pts,
                               const float* __restrict__ w2v,
                               u32* __restrict__ keys, u32* __restrict__ idxs, int n) {
  __shared__ __align__(16) float lpts[256 * 3];
  const int base = blockIdx.x * 256;
#if USE_TDM
  if (threadIdx.x < 32) {  // one wave issues the DMA (EXEC ignored by TDM)
    tdm_load_dwords(lpts, pts + (size_t)base * 3, (u32)(n - base) * 3u, 768u);
    __builtin_amdgcn_s_wait_tensorcnt((short)0);  // per-wave TENSORcnt
  }
  __syncthreads();
  const int i = base + (int)threadIdx.x;
  if (i < n) {
    const float x = lpts[threadIdx.x * 3 + 0];
    const float y = lpts[threadIdx.x * 3 + 1];
    const float z = lpts[threadIdx.x * 3 + 2];
#else
  const int i = base + (int)threadIdx.x;
  if (i < n) {
    const float x = pts[i * 3 + 0];
    const float y = pts[i * 3 + 1];
    const float z = pts[i * 3 + 2];
#endif
    const float zv = x * w2v[2] + y * w2v[6] + z * w2v[10] + w2v[14];
    const float key = (zv >= 0.2f) ? zv : __builtin_inff();
    keys[i] = key_bits(key);
    idxs[i] = (u32)i;
  }
}

// -------------------------------------------------------------------------
// Radix sort: 4-bit digits, 8 stable LSD passes over (key, idx) pairs.
// Tile loads (contiguous 16/32 KB) are DMAed into LDS by the TDM.
// -------------------------------------------------------------------------
__global__ void gs_hist_kernel(const u32* __restrict__ keys, u32* __restrict__ hist,
                               int shift, int n, int nb) {
  __shared__ u32 cnt[RADIX];
  __shared__ __align__(16) u32 lkeys[SORT_TILE];  // 16 KB
  const int base = blockIdx.x * SORT_TILE;
  if (threadIdx.x < RADIX) cnt[threadIdx.x] = 0;
#if USE_TDM
  if (threadIdx.x < 32) {
    tdm_load_dwords(lkeys, keys + base, (u32)(n - base), SORT_TILE);
    __builtin_amdgcn_s_wait_tensorcnt((short)0);
  }
#endif
  __syncthreads();
  for (int k = threadIdx.x; k < SORT_TILE; k += SORT_THREADS) {
    const int i = base + k;
    if (i < n) {
#if USE_TDM
      const u32 kv = lkeys[k];
#else
      const u32 kv = keys[i];
#endif
      atomicAdd(&cnt[(kv >> shift) & (RADIX - 1)], 1u);
    }
  }
  __syncthreads();
  // digit-major layout: a single global exclusive scan yields (digit, block) bases
  if (threadIdx.x < RADIX) hist[threadIdx.x * nb + blockIdx.x] = cnt[threadIdx.x];
}

__global__ void gs_scan_kernel(u32* __restrict__ data, int L) {
  __shared__ u32 tmp[SORT_THREADS];
  __shared__ u32 carry;
  const int t = threadIdx.x;
  if (t == 0) carry = 0;
  __syncthreads();
  for (int base = 0; base < L; base += SORT_THREADS) {
    const u32 c = carry;
    const int i = base + t;
    const u32 v = (i < L) ? data[i] : 0u;
    tmp[t] = v;
    __syncthreads();
    for (int off = 1; off < SORT_THREADS; off <<= 1) {
      const u32 a = (t >= off) ? tmp[t - off] : 0u;
      __syncthreads();
      tmp[t] += a;
      __syncthreads();
    }
    if (i < L) data[i] = c + tmp[t] - v;  // exclusive
    __syncthreads();
    if (t == 0) carry = c + tmp[SORT_THREADS - 1];
    __syncthreads();
  }
}

__global__ void gs_scatter_kernel(const u32* __restrict__ keys_in, const u32* __restrict__ idx_in,
                                  u32* __restrict__ keys_out, u32* __restrict__ idx_out,
                                  const u32* __restrict__ offs, int shift, int n, int nb) {
  __shared__ u32 cnt[RADIX * SORT_THREADS];       // 16 KB
  __shared__ u32 base_s[RADIX];
#if USE_TDM
  __shared__ __align__(16) u32 lkeys[SORT_TILE];  // 16 KB
  __shared__ __align__(16) u32 lidx[SORT_TILE];   // 16 KB
#endif
  const int t = threadIdx.x;
  const int tilebase = blockIdx.x * SORT_TILE;
#pragma unroll
  for (int d = 0; d < RADIX; ++d) cnt[d * SORT_THREADS + t] = 0;
#if USE_TDM
  if (t < 32) {
    tdm_load_dwords(lkeys, keys_in + tilebase, (u32)(n - tilebase), SORT_TILE);
    tdm_load_dwords(lidx, idx_in + tilebase, (u32)(n - tilebase), SORT_TILE);
    __builtin_amdgcn_s_wait_tensorcnt((short)0);  // both DMAs complete
  }
#endif
  __syncthreads();
  // Each thread owns SORT_ITEMS consecutive elements -> stable element order.
  const int local = t * SORT_ITEMS;
  u32 k[SORT_ITEMS], id[SORT_ITEMS], rk[SORT_ITEMS], dg[SORT_ITEMS];
#pragma unroll
  for (int j = 0; j < SORT_ITEMS; ++j) {
    const int i = tilebase + local + j;
    if (i < n) {
#if USE_TDM
      k[j] = lkeys[local + j];
      id[j] = lidx[local + j];
#else
      k[j] = keys_in[i];
      id[j] = idx_in[i];
#endif
      dg[j] = (k[j] >> shift) & (RADIX - 1);
      const u32 c = cnt[dg[j] * SORT_THREADS + t];
      rk[j] = c;
      cnt[dg[j] * SORT_THREADS + t] = c + 1u;
    }
  }
  __syncthreads();
  // exclusive prefix across threads per digit (serial per digit-row: tiny, hidden by occupancy)
  if (t < RADIX) {
    u32 run = 0;
    for (int j = 0; j < SORT_THREADS; ++j) {
      const u32 v = cnt[t * SORT_THREADS + j];
      cnt[t * SORT_THREADS + j] = run;
      run += v;
    }
    base_s[t] = offs[t * nb + blockIdx.x];
  }
  __syncthreads();
#pragma unroll
  for (int j = 0; j < SORT_ITEMS; ++j) {
    const int i = tilebase + local + j;
    if (i < n) {
      const u32 pos = base_s[dg[j]] + cnt[dg[j] * SORT_THREADS + t] + rk[j];
      keys_out[pos] = k[j];
      idx_out[pos] = id[j];
    }
  }
}

// -------------------------------------------------------------------------
// Pass 3: gather + full transform, LDS-staged coalesced float4 output.
// -------------------------------------------------------------------------
__global__ void gs_gather_kernel(const u32* __restrict__ order,
                                 const float* __restrict__ pts, const float* __restrict__ cols,
                                 const float* __restrict__ opac, const float* __restrict__ scl,
                                 const float* __restrict__ quat,
                                 const float* __restrict__ w2v, const float* __restrict__ fullm,
                                 const float* __restrict__ tanx_p, const float* __restrict__ tany_p,
                                 const float* __restrict__ fx_p, const float* __restrict__ fy_p,
                                 const int* __restrict__ wid_p, const int* __restrict__ hei_p,
                                 float* __restrict__ out, int n) {
  __shared__ __align__(16) float row[256 * 21];  // 21 KB staging
  const int t = threadIdx.x;
  const int r = blockIdx.x * 256 + t;
  if (r < n) {
    const int src = (int)order[r];
    const float tanx = tanx_p[0], tany = tany_p[0];
    const float fx = fx_p[0], fy = fy_p[0];
    const float Wf = (float)wid_p[0], Hf = (float)hei_p[0];
    // quaternion -> rotation
    const float4 q = reinterpret_cast<const float4*>(quat)[src];
    const float qn = 1.0f / sqrtf(q.x * q.x + q.y * q.y + q.z * q.z + q.w * q.w);
    const float qw = q.x * qn, qx = q.y * qn, qy = q.z * qn, qz = q.w * qn;
    const float R00 = 1.f - 2.f * (qy * qy + qz * qz), R01 = 2.f * (qx * qy - qw * qz), R02 = 2.f * (qx * qz + qw * qy);
    const float R10 = 2.f * (qx * qy + qw * qz), R11 = 1.f - 2.f * (qx * qx + qz * qz), R12 = 2.f * (qy * qz - qw * qx);
    const float R20 = 2.f * (qx * qz - qw * qy), R21 = 2.f * (qy * qz + qw * qx), R22 = 1.f - 2.f * (qx * qx + qy * qy);
    const float s0 = scl[src * 3 + 0], s1 = scl[src * 3 + 1], s2 = scl[src * 3 + 2];
    const float M00 = R00 * s0, M01 = R01 * s1, M02 = R02 * s2;
    const float M10 = R10 * s0, M11 = R11 * s1, M12 = R12 * s2;
    const float M20 = R20 * s0, M21 = R21 * s1, M22 = R22 * s2;
    const float c3_00 = M00 * M00 + M01 * M01 + M02 * M02;
    const float c3_01 = M00 * M10 + M01 * M11 + M02 * M12;
    const float c3_02 = M00 * M20 + M01 * M21 + M02 * M22;
    const float c3_11 = M10 * M10 + M11 * M11 + M12 * M12;
    const float c3_12 = M10 * M20 + M11 * M21 + M12 * M22;
    const float c3_22 = M20 * M20 + M21 * M21 + M22 * M22;
    // projection (row-vector @ matrix convention)
    const float px = pts[src * 3 + 0], py = pts[src * 3 + 1], pz = pts[src * 3 + 2];
    const float pv0 = px * w2v[0] + py * w2v[4] + pz * w2v[8] + w2v[12];
    const float pv1 = px * w2v[1] + py * w2v[5] + pz * w2v[9] + w2v[13];
    const float zv  = px * w2v[2] + py * w2v[6] + pz * w2v[10] + w2v[14];
    const float pc0 = px * fullm[0] + py * fullm[4] + pz * fullm[8] + fullm[12];
    const float pc1 = px * fullm[1] + py * fullm[5] + pz * fullm[9] + fullm[13];
    const float pc3 = px * fullm[3] + py * fullm[7] + pz * fullm[11] + fullm[15];
    const float inv_w = 1.0f / pc3;
    const float sx = ((pc0 * inv_w + 1.0f) * Wf - 1.0f) * 0.5f;
    const float sy = ((pc1 * inv_w + 1.0f) * Hf - 1.0f) * 0.5f;
    const float lx = 1.3f * tanx, ly = 1.3f * tany;
    const float xc = fminf(fmaxf(pv0 / zv, -lx), lx) * zv;
    const float yc = fminf(fmaxf(pv1 / zv, -ly), ly) * zv;
    const float z2 = zv * zv;
    const float J00 = fx / zv, J02 = -(fx * xc) / z2;
    const float J11 = fy / zv, J12 = -(fy * yc) / z2;
    // T = J @ W^T where W^T[k][j] = w2v[j*4+k]
    const float T00 = J00 * w2v[0] + J02 * w2v[2];
    const float T01 = J00 * w2v[4] + J02 * w2v[6];
    const float T02 = J00 * w2v[8] + J02 * w2v[10];
    const float T10 = J11 * w2v[1] + J12 * w2v[2];
    const float T11 = J11 * w2v[5] + J12 * w2v[6];
    const float T12 = J11 * w2v[9] + J12 * w2v[10];
    const float u0 = c3_00 * T00 + c3_01 * T01 + c3_02 * T02;
    const float u1 = c3_01 * T00 + c3_11 * T01 + c3_12 * T02;
    const float u2 = c3_02 * T00 + c3_12 * T01 + c3_22 * T02;
    const float v0 = c3_00 * T10 + c3_01 * T11 + c3_02 * T12;
    const float v1 = c3_01 * T10 + c3_11 * T11 + c3_12 * T12;
    const float v2 = c3_02 * T10 + c3_12 * T11 + c3_22 * T12;
    const float c00 = T00 * u0 + T01 * u1 + T02 * u2;
    const float c01 = T00 * v0 + T01 * v1 + T02 * v2;
    const float c10 = T10 * u0 + T11 * u1 + T12 * u2;
    const float c11 = T10 * v0 + T11 * v1 + T12 * v2;
    const float det = c00 * c11 - c01 * c10;
    const float det_safe = (fabsf(det) < 1e-6f) ? 1e-6f : det;
    const float inv_det = 1.0f / det_safe;
    const float mid = 0.5f * (c00 + c11);
    const float inter = fmaxf(mid * mid - det, 0.1f);
    const float lam = mid + sqrtf(inter);
    const float radius = ceilf(3.0f * sqrtf(fmaxf(lam, 0.0f)));
    float* o = &row[t * 21];
    o[0] = sx; o[1] = sy; o[2] = zv;
    o[3] = c00; o[4] = c01; o[5] = c10; o[6] = c11;
    o[7] = c11 * inv_det; o[8] = -c01 * inv_det; o[9] = -c10 * inv_det; o[10] = c00 * inv_det;
    o[11] = radius;
    o[12] = floorf(sx - radius); o[13] = floorf(sy - radius);
    o[14] = ceilf(sx + radius);  o[15] = ceilf(sy + radius);
    o[16] = cols[src * 3 + 0]; o[17] = cols[src * 3 + 1]; o[18] = cols[src * 3 + 2];
    o[19] = 1.0f / (1.0f + expf(-opac[src]));
    o[20] = (zv >= 0.2f) ? 1.0f : 0.0f;
  }
  __syncthreads();
  const int nrows = (n - blockIdx.x * 256 < 256) ? (n - blockIdx.x * 256) : 256;
  float* dst = out + (size_t)blockIdx.x * (256 * 21);
  if (nrows == 256) {
    float4* d4 = reinterpret_cast<float4*>(dst);           // block base offset 21504 B: 16B-aligned
    const float4* s4 = reinterpret_cast<const float4*>(row);
    for (int j = t; j < (256 * 21) / 4; j += 256) d4[j] = s4[j];
  } else {
    for (int j = t; j < nrows * 21; j += 256) dst[j] = row[j];
  }
}

extern "C" void kernel_launch(void* const* d_in, const int* in_sizes, int n_in,
                              void* d_out, int out_size, void* d_ws, size_t ws_size,
                              hipStream_t stream) {
  (void)n_in; (void)out_size; (void)ws_size;
  const int n = in_sizes[2];  // opacity count == N
  const float* pts  = (const float*)d_in[0];
  const float* cols = (const float*)d_in[1];
  const float* opac = (const float*)d_in[2];
  const float* scl  = (const float*)d_in[3];
  const float* quat = (const float*)d_in[4];
  const float* w2v  = (const float*)d_in[5];
  const float* fullm = (const float*)d_in[6];
  const float* tanx = (const float*)d_in[7];
  const float* tany = (const float*)d_in[8];
  const float* fx   = (const float*)d_in[9];
  const float* fy   = (const float*)d_in[10];
  const int* wid    = (const int*)d_in[11];
  const int* hei    = (const int*)d_in[12];
  float* out = (float*)d_out;

  // workspace: keysA|idxA|keysB|idxB|hist  (~32 MB for N=2M)
  u32* keysA = (u32*)d_ws;
  u32* idxA  = keysA + n;
  u32* keysB = idxA + n;
  u32* idxB  = keysB + n;
  u32* hist  = idxB + n;

  const int nb = (n + SORT_TILE - 1) / SORT_TILE;
  const int nthread_blocks = (n + 255) / 256;

  gs_keys_kernel<<<nthread_blocks, 256, 0, stream>>>(pts, w2v, keysA, idxA, n);

  u32 *kin = keysA, *iin = idxA, *kout = keysB, *iout = idxB;
  for (int pass = 0; pass < 8; ++pass) {
    const int shift = pass * 4;
    gs_hist_kernel<<<nb, SORT_THREADS, 0, stream>>>(kin, hist, shift, n, nb);
    gs_scan_kernel<<<1, SORT_THREADS, 0, stream>>>(hist, RADIX * nb);
    gs_scatter_kernel<<<nb, SORT_THREADS, 0, stream>>>(kin, iin, kout, iout, hist, shift, n, nb);
    u32* tk = kin; kin = kout; kout = tk;
    u32* ti = iin; iin = iout; iout = ti;
  }
  // 8 passes (even) -> sorted result back in keysA/idxA == kin/iin

  gs_gather_kernel<<<nthread_blocks, 256, 0, stream>>>(
      iin, pts, cols, opac, scl, quat, w2v, fullm,
      tanx, tany, fx, fy, wid, hei, out, n);
}